// VectorQuantizer_64561948393896
// MI455X (gfx1250) — compile-verified
//
#include <hip/hip_runtime.h>
#include <stdint.h>

#define BS 65536
#define D  768
#define NTOT ((long long)BS * (long long)D)   // 50,331,648 elements
#define N4   ((int)(NTOT / 4))                 // 12,582,912 float4 (fits int)
#define BLOCK 256
#define ITER 8
#define PER_BLOCK (BLOCK * ITER)               // 2048 float4 per block
#define NBLOCKS (N4 / PER_BLOCK)               // 6144 blocks, exact

typedef __attribute__((ext_vector_type(4))) float v4f;

__global__ void vq_zero_acc(double* acc) { *acc = 0.0; }

__global__ __launch_bounds__(BLOCK) void vq_main(
    const float* __restrict__ x, const int* __restrict__ cat,
    const float* __restrict__ cb, float* __restrict__ out,
    double* __restrict__ acc)
{
    __shared__ v4f buf[2][BLOCK];       // double-buffered async staging, 8 KB
    __shared__ float wsum[BLOCK / 32];

    const int tid  = threadIdx.x;
    const int base = (int)blockIdx.x * PER_BLOCK;   // max 12.6M, fits int
    const v4f* __restrict__ xg  = (const v4f*)x;
    const v4f* __restrict__ cbg = (const v4f*)cb;
    v4f* __restrict__ outg      = (v4f*)out;

    // LDS byte offsets of this thread's private 16B slots in each buffer
    uint32_t lds0 = (uint32_t)(uintptr_t)&buf[0][tid];
    uint32_t lds1 = (uint32_t)(uintptr_t)&buf[1][tid];

    // Prefetch chunk 0 (async VMEM -> LDS, tracked by ASYNCcnt)
    {
        const v4f* gp = xg + (base + tid);
        asm volatile("global_load_async_to_lds_b128 %0, %1, off"
                     :: "v"(lds0), "v"(gp) : "memory");
    }

    float accf = 0.0f;
#pragma unroll
    for (int i = 0; i < ITER; ++i) {
        if (i + 1 < ITER) {
            // Prefetch chunk i+1 into the other buffer, then wait so that
            // only that one is outstanding -> chunk i has landed (in-order).
            const v4f* gp = xg + (base + (i + 1) * BLOCK + tid);
            uint32_t l = ((i + 1) & 1) ? lds1 : lds0;
            asm volatile("global_load_async_to_lds_b128 %0, %1, off"
                         :: "v"(l), "v"(gp) : "memory");
            asm volatile("s_wait_asynccnt 0x1" ::: "memory");
        } else {
            asm volatile("s_wait_asynccnt 0x0" ::: "memory");
        }

        const int idx4 = base + i * BLOCK + tid;   // 32-bit: enables saddr+voffset
        const int row  = idx4 / 192;               // 192 float4 per row
        const int col  = idx4 - row * 192;
        const int c    = cat[row];                 // cached, 0.26 MB total
        const v4f q4   = cbg[c * 192 + col];       // gather, L2-resident (1.5 MB)
        const v4f x4   = buf[i & 1][tid];          // staged via async

        const v4f d4 = q4 - x4;        // diff (matches reference FP op order)
        const v4f o4 = x4 + d4;        // straight-through output
        __builtin_nontemporal_store(o4, outg + idx4);   // 201 MB stream, spare L2

        accf += d4.x * d4.x + d4.y * d4.y + d4.z * d4.z + d4.w * d4.w;
    }

    // wave32 reduction, then block reduction, then one f64 atomic per block
    float s = accf;
    for (int o = 16; o > 0; o >>= 1) s += __shfl_down(s, o, 32);
    const int lane = tid & 31, wave = tid >> 5;
    if (lane == 0) wsum[wave] = s;
    __syncthreads();
    if (tid == 0) {
        float b = 0.0f;
        for (int w = 0; w < BLOCK / 32; ++w) b += wsum[w];
        atomicAdd(acc, (double)b);     // global_atomic_add_f64
    }
}

__global__ void vq_finalize(const double* __restrict__ acc, float* __restrict__ out) {
    const double mse = *acc / (double)NTOT;
    // loss = CODEBOOK_COST * mse + COMMITMENT_COST * mse = 1.25 * mse
    out[NTOT] = (float)(1.25 * mse);
}

extern "C" void kernel_launch(void* const* d_in, const int* in_sizes, int n_in,
                              void* d_out, int out_size, void* d_ws, size_t ws_size,
                              hipStream_t stream) {
    const float* x   = (const float*)d_in[0];   // inputs   [BS, D] f32
    const int*   cat = (const int*)d_in[1];     // categories [BS] int
    const float* cb  = (const float*)d_in[2];   // codebook [NE, D] f32
    float* out  = (float*)d_out;                // [BS*D] quantized_st + [1] loss
    double* acc = (double*)d_ws;                // f64 accumulator in scratch

    vq_zero_acc<<<1, 1, 0, stream>>>(acc);
    vq_main<<<NBLOCKS, BLOCK, 0, stream>>>(x, cat, cb, out, acc);
    vq_finalize<<<1, 1, 0, stream>>>(acc, out);
}